// FocalCTCLoss_85074712199475
// MI455X (gfx1250) — compile-verified
//
#include <hip/hip_runtime.h>
#include <hip/hip_bf16.h>
#include <stdint.h>

// Problem constants (from setup_inputs): B=32, C=128, T=2048, S=128
#define B_ 32
#define C_ 128
#define T_ 2048
#define S_ 128
#define L_ 257          // 2*S+1 extended CTC states
#define GAMMA_ 2.0f
#define NEG_ -1e30f
#define CH_ 8           // async emit-staging chunk (time steps)

// gfx1250 async Global->LDS path (gate on toolchain support; sync fallback).
#if defined(__has_builtin)
# if __has_builtin(__builtin_amdgcn_global_load_async_to_lds_b32) && \
     __has_builtin(__builtin_amdgcn_s_wait_asynccnt)
#  define HAVE_ASYNC 1
# endif
#endif
#ifndef HAVE_ASYNC
# define HAVE_ASYNC 0
#endif

typedef float v2f __attribute__((ext_vector_type(2)));
typedef float v8f __attribute__((ext_vector_type(8)));

__device__ __forceinline__ float lae2(float a, float b) {
    float m = fmaxf(a, b);
    return m + __logf(__expf(a - m) + __expf(b - m));
}

#if HAVE_ASYNC
// One 32-bit per-lane async copy Global -> LDS (tracked by ASYNCcnt).
// Builtin parameter types per hipcc diagnostic: (as1 int*, as3 int*, Ii, Ii).
__device__ __forceinline__ void async_ld_f32(const float* g, float* lds) {
    __builtin_amdgcn_global_load_async_to_lds_b32(
        (__attribute__((address_space(1))) int*)const_cast<float*>(g),
        (__attribute__((address_space(3))) int*)lds, 0, 0);
}
#endif

// ---------------------------------------------------------------------------
// Kernel 1: log-softmax over the class axis of logits (B, C, T).
// Sum-of-exp over C=128 runs on the matrix pipe with V_WMMA_F32_16X16X4_F32:
//   A (16x4 f32)  : rows = 16 t's, K = 4 classes of exp(x - max)
//   B (4x16 ones) : every column of D becomes the running row-sum
// 32 chained WMMAs cover K=128 at full fp32 precision.
// ---------------------------------------------------------------------------
__global__ __launch_bounds__(256) void k_logsoftmax(const float* __restrict__ logits,
                                                    float* __restrict__ lp)
{
    const int wave  = threadIdx.x >> 5;
    const int lane  = threadIdx.x & 31;
    const int m     = lane & 15;        // row (t) index within tile
    const int khalf = lane >> 4;        // which K-half this lane supplies
    const int tilesPerB = T_ / 128;     // 16 tiles of 128 t per batch
    const int b  = blockIdx.x / tilesPerB;
    const int t  = (blockIdx.x % tilesPerB) * 128 + wave * 16 + m;

    const float* __restrict__ row = logits + (size_t)b * C_ * T_;

    // pass 1: max over classes (lane covers half the classes, share via xor-16)
    float mx = -3.4e38f;
    for (int c = khalf; c < C_; c += 2)
        mx = fmaxf(mx, row[(size_t)c * T_ + t]);
    mx = fmaxf(mx, __shfl_xor(mx, 16, 32));

    // pass 2: sum of exp via fp32 WMMA row-sums against an all-ones B matrix
    v8f acc = {};
    v2f bones; bones[0] = 1.0f; bones[1] = 1.0f;
    #pragma unroll 4
    for (int it = 0; it < 32; ++it) {
        int cb = it * 4 + khalf * 2;    // lanes 0-15: K=0,1  lanes 16-31: K=2,3
        v2f a;
        a[0] = __expf(row[(size_t)cb * T_ + t] - mx);
        a[1] = __expf(row[(size_t)(cb + 1) * T_ + t] - mx);
        acc = __builtin_amdgcn_wmma_f32_16x16x4_f32(false, a, false, bones,
                                                    (short)0, acc, false, false);
    }
    // D layout: lanes 0-15 hold rows M=0..7 in vgprs 0..7, lanes 16-31 rows 8..15
    int msel = m & 7;
    float sv = acc[0];
    #pragma unroll
    for (int r = 1; r < 8; ++r) sv = (msel == r) ? acc[r] : sv;
    float so = __shfl_xor(sv, 16, 32);
    float sumexp = (khalf == (m >> 3)) ? sv : so;
    float lse = mx + __logf(sumexp);

    // pass 3: write lp = x - max - log(sum)
    float* __restrict__ orow = lp + (size_t)b * C_ * T_;
    for (int cc = khalf * 64; cc < khalf * 64 + 64; ++cc)
        orow[(size_t)cc * T_ + t] = row[(size_t)cc * T_ + t] - lse;
}

// ---------------------------------------------------------------------------
// Kernel 2: CTC forward scan. One workgroup per batch element, 288 threads
// (9 wave32), thread l owns extended state l (l < 257). Alpha lives in LDS.
// Emit values are streamed through LDS with double-buffered per-lane
// GLOBAL_LOAD_ASYNC_TO_LDS_B32: the next 8-step chunk is fetched while the
// current one feeds the recurrence; s_wait_asynccnt 8 fences only the
// previous chunk (per-wave counter, loads complete in order, each lane
// consumes exactly the slots its own async loads wrote -> no barriers).
// ---------------------------------------------------------------------------
__global__ __launch_bounds__(288) void k_ctc_forward(
    const float* __restrict__ lp, const int* __restrict__ targets,
    const int* __restrict__ input_lens, const int* __restrict__ target_lens,
    float* __restrict__ alphaWS, float* __restrict__ lossb,
    float* __restrict__ logZb, int* __restrict__ validb)
{
    __shared__ float sA[L_ + 2];    // sA[0..1] = NEG pads, state l at sA[l+2]
#if HAVE_ASYNC
    __shared__ float sEm[2][CH_][L_ + 3];   // double-buffered emit staging
#endif
    const int b = blockIdx.x;
    const int l = threadIdx.x;
    const bool act = l < L_;
    const int* tgt = targets + b * S_;

    int myc = (act && (l & 1)) ? tgt[(l - 1) >> 1] : 0;
    bool allow2 = false;
    if (act && l >= 3 && myc != 0) {            // only odd l>=3 can skip
        int cpp = tgt[(l - 3) >> 1];
        allow2 = (myc != cpp);
    }
    const float* rowp = lp + (size_t)b * C_ * T_ + (size_t)myc * T_;
    float* aW = alphaWS + (size_t)b * T_ * L_;

    float a0 = act ? ((l < 2) ? rowp[0] : NEG_) : NEG_;
    if (threadIdx.x < 2) sA[threadIdx.x] = NEG_;
    if (act) { sA[l + 2] = a0; aW[l] = a0; }
    const int il = min(input_lens[b], T_);
    __syncthreads();

#if HAVE_ASYNC
    // prime chunk 0 (t = 1 .. CH_)
    if (act) {
        #pragma unroll
        for (int j = 0; j < CH_; ++j)
            async_ld_f32(rowp + 1 + j, &sEm[0][j][l]);
    }
#endif
    for (int n = 0; n * CH_ + 1 < T_; ++n) {
        const int tb = 1 + n * CH_;
        const int cnt = min(CH_, T_ - tb);
#if HAVE_ASYNC
        const int tbn = tb + CH_;
        const int cnext = (tbn < T_) ? min(CH_, T_ - tbn) : 0;
        if (act) {
            for (int j = 0; j < cnext; ++j)
                async_ld_f32(rowp + tbn + j, &sEm[(n + 1) & 1][j][l]);
        }
        if (cnext == CH_) __builtin_amdgcn_s_wait_asynccnt(CH_);
        else              __builtin_amdgcn_s_wait_asynccnt(0);
#endif
        for (int j = 0; j < cnt; ++j) {
            const int t = tb + j;
#if HAVE_ASYNC
            float e = act ? sEm[n & 1][j][l] : NEG_;
#else
            float e = act ? rowp[t] : NEG_;
#endif
            int li = act ? l : 0;
            float am = sA[li + 2];
            float a1 = sA[li + 1];
            float a2 = allow2 ? sA[li] : NEG_;
            float nv = lae2(lae2(am, a1), a2) + e;
            if (t >= il) nv = am;               // mask past input length
            __syncthreads();
            if (act) sA[l + 2] = nv;
            __syncthreads();
            if (act) aW[(size_t)t * L_ + l] = nv;
        }
    }
    if (threadIdx.x == 0) {
        int end = 2 * target_lens[b];
        float tot = lae2(sA[end + 2], sA[end + 1]);
        int vld = tot > 0.5f * NEG_;
        lossb[b]  = vld ? -tot : 0.0f;          // zero_infinity
        logZb[b]  = tot;
        validb[b] = vld;
    }
}

// ---------------------------------------------------------------------------
// Kernel 3: CTC backward scan + gradient gather. Beta in LDS; at each t the
// posterior w[l]=exp(alpha+beta-logZ) is gathered per class into LDS
// (ds_add_f32 atomics for labels, deterministic wave reductions for blanks
// and the per-(b,t) total S). gam stored in (B,C,T) layout to match lp.
// ---------------------------------------------------------------------------
__global__ __launch_bounds__(288) void k_ctc_backward(
    const float* __restrict__ lp, const int* __restrict__ targets,
    const int* __restrict__ target_lens,
    const float* __restrict__ alphaWS, const float* __restrict__ logZb,
    const int* __restrict__ validb,
    float* __restrict__ gam, float* __restrict__ Sbt)
{
    __shared__ float sB[L_ + 2];    // beta[t];  pads at [L_], [L_+1]
    __shared__ float sBh[L_ + 2];   // beta + emit, same pads
    __shared__ float gcls[C_];
    __shared__ float swave[16];
    __shared__ float bwave[16];
    const int b = blockIdx.x;
    const int l = threadIdx.x;
    const bool act = l < L_;
    const int lane = l & 31, wid = l >> 5;      // 9 waves
    const int* tgt = targets + b * S_;

    int myc = (act && (l & 1)) ? tgt[(l - 1) >> 1] : 0;
    bool allowUp = false;                       // transition l -> l+2 allowed?
    if (act && (l & 1) && (l + 2) < L_) {
        int cup = tgt[(l + 1) >> 1];
        allowUp = (cup != 0) && (cup != myc);
    }
    const float* rowp = lp + (size_t)b * C_ * T_ + (size_t)myc * T_;
    const float* aW = alphaWS + (size_t)b * T_ * L_;
    const float lz = logZb[b];
    const int vld = validb[b];
    int end = 2 * target_lens[b];

    if (act) sB[l] = (l == end || l == end - 1) ? 0.0f : NEG_;
    if (l == 0) { sB[L_] = NEG_; sB[L_ + 1] = NEG_; sBh[L_] = NEG_; sBh[L_ + 1] = NEG_; }
    __syncthreads();

    for (int t = T_ - 1; t >= 0; --t) {
        int li = act ? l : 0;
        float w = 0.0f;
        if (act && vld) {
            float e = aW[(size_t)t * L_ + l] + sB[l] - lz;
            w = (e > -60.0f) ? __expf(e) : 0.0f;
        }
        if (threadIdx.x < C_) gcls[threadIdx.x] = 0.0f;
        __syncthreads();
        // per-wave reductions: total posterior + blank class (deterministic)
        float wl = w;
        float wb = (myc == 0) ? w : 0.0f;
        #pragma unroll
        for (int off = 16; off > 0; off >>= 1) {
            wl += __shfl_xor(wl, off, 32);
            wb += __shfl_xor(wb, off, 32);
        }
        if (lane == 0) { swave[wid] = wl; bwave[wid] = wb; }
        if (act && myc != 0) atomicAdd(&gcls[myc], w);   // ds_add_f32
        __syncthreads();
        if (threadIdx.x == 0) {
            float st = 0.0f, bl = 0.0f;
            for (int i = 0; i < 9; ++i) { st += swave[i]; bl += bwave[i]; }
            gcls[0] = bl;                                 // labels are never 0
            Sbt[(size_t)b * T_ + t] = st;
        }
        __syncthreads();
        if (threadIdx.x < C_)
            gam[(size_t)b * C_ * T_ + (size_t)threadIdx.x * T_ + t] = gcls[threadIdx.x];
        if (t > 0) {
            float bh = act ? (sB[l] + rowp[t]) : NEG_;
            if (act) __builtin_prefetch(rowp + t - 32, 0, 0);
            if (act) sBh[l] = bh;
            __syncthreads();
            float nb = lae2(lae2(sBh[li], sBh[li + 1]),
                            allowUp ? sBh[li + 2] : NEG_);
            __syncthreads();
            if (act) sB[l] = nb;
            __syncthreads();
        }
    }
}

// ---------------------------------------------------------------------------
// Kernel 4: elementwise focal term + per-block partial sums.
//   grads  = p*S - gam      (d vanilla / d logits)
//   gct    = p - grads
//   focal  = max(1-p,1e-5)^2
//   contrib= focal * gct * lp
// ---------------------------------------------------------------------------
__global__ __launch_bounds__(256) void k_focal_partial(
    const float* __restrict__ lp, const float* __restrict__ gam,
    const float* __restrict__ Sbt, float* __restrict__ partials)
{
    const int tid = threadIdx.x;
    size_t base = (size_t)blockIdx.x * 2048 + tid;
    float acc = 0.0f;
    #pragma unroll
    for (int i = 0; i < 8; ++i) {
        size_t idx = base + (size_t)i * 256;
        float lv = lp[idx];
        float g  = gam[idx];
        int bb = (int)(idx >> 18);          // C_*T_ = 2^18
        int tt = (int)(idx & (T_ - 1));
        float Sv = Sbt[(size_t)bb * T_ + tt];
        float p = __expf(lv);
        float grads = p * Sv - g;
        float gct = p - grads;
        float om = fmaxf(1.0f - p, 1e-5f);
        acc += om * om * gct * lv;          // GAMMA == 2 -> square
    }
    #pragma unroll
    for (int off = 16; off > 0; off >>= 1) acc += __shfl_xor(acc, off, 32);
    __shared__ float wsum[8];
    if ((tid & 31) == 0) wsum[tid >> 5] = acc;
    __syncthreads();
    if (tid == 0) {
        float s = 0.0f;
        for (int i = 0; i < 8; ++i) s += wsum[i];
        partials[blockIdx.x] = s;
    }
}

// ---------------------------------------------------------------------------
// Kernel 5: final scalar.
// ---------------------------------------------------------------------------
__global__ __launch_bounds__(256) void k_final(
    const float* __restrict__ partials, int nPart,
    const float* __restrict__ lossb, const int* __restrict__ target_lens,
    float* __restrict__ out)
{
    const int tid = threadIdx.x;
    float s = 0.0f;
    for (int i = tid; i < nPart; i += 256) s += partials[i];
    #pragma unroll
    for (int off = 16; off > 0; off >>= 1) s += __shfl_xor(s, off, 32);
    __shared__ float wsum[8];
    if ((tid & 31) == 0) wsum[tid >> 5] = s;
    __syncthreads();
    if (tid == 0) {
        float wx = 0.0f;
        for (int i = 0; i < 8; ++i) wx += wsum[i];
        float vanilla = 0.0f; int dsum = 0;
        for (int b = 0; b < B_; ++b) { vanilla += lossb[b]; dsum += target_lens[b]; }
        float denom = (float)max(dsum, 1);
        float weighted_xent = -wx;
        out[0] = (GAMMA_ * weighted_xent + (1.0f - GAMMA_) * vanilla) / denom;
    }
}

// ---------------------------------------------------------------------------
extern "C" void kernel_launch(void* const* d_in, const int* in_sizes, int n_in,
                              void* d_out, int out_size, void* d_ws, size_t ws_size,
                              hipStream_t stream)
{
    (void)in_sizes; (void)n_in; (void)out_size; (void)ws_size;
    const float* logits      = (const float*)d_in[0];
    const int*   targets     = (const int*)d_in[1];
    const int*   input_lens  = (const int*)d_in[2];
    const int*   target_lens = (const int*)d_in[3];
    float* out = (float*)d_out;

    char* ws = (char*)d_ws;
    const size_t LPB = (size_t)B_ * C_ * T_ * sizeof(float);   // 33.5 MB
    const size_t ALB = (size_t)B_ * T_ * L_ * sizeof(float);   // 67.4 MB
    float* lp      = (float*)(ws);
    float* gam     = (float*)(ws + LPB);
    float* alphaWS = (float*)(ws + 2 * LPB);
    float* Sbt     = (float*)(ws + 2 * LPB + ALB);
    char*  tail    = ws + 2 * LPB + ALB + (size_t)B_ * T_ * sizeof(float);
    float* lossb    = (float*)(tail);
    float* logZb    = (float*)(tail + 256);
    int*   validb   = (int*)  (tail + 512);
    float* partials = (float*)(tail + 1024);
    const int nPart = (B_ * C_ * T_) / 2048;                   // 4096

    k_logsoftmax<<<dim3(B_ * (T_ / 128)), dim3(256), 0, stream>>>(logits, lp);
    k_ctc_forward<<<dim3(B_), dim3(288), 0, stream>>>(lp, targets, input_lens,
                                                      target_lens, alphaWS,
                                                      lossb, logZb, validb);
    k_ctc_backward<<<dim3(B_), dim3(288), 0, stream>>>(lp, targets, target_lens,
                                                       alphaWS, logZb, validb,
                                                       gam, Sbt);
    k_focal_partial<<<dim3(nPart), dim3(256), 0, stream>>>(lp, gam, Sbt, partials);
    k_final<<<dim3(1), dim3(256), 0, stream>>>(partials, nPart, lossb,
                                               target_lens, out);
}